// ShortConv1d_35545149342129
// MI455X (gfx1250) — compile-verified
//
#include <hip/hip_runtime.h>
#include <stdint.h>

// Problem constants (from reference): B=8, S=4096, D=2048, K=4
#define B_ 8
#define S_ 4096
#define D_ 2048

// Tiling: each block = 128 threads (4 wave32), owns 512 channels (float4/thread),
// streams CH=256 sequence rows in NS=32 async double-buffered stages of TS=8 rows.
#define CH 256
#define TS 8
#define NS (CH / TS)
#define DT 512
#define THREADS (DT / 4)

__global__ __launch_bounds__(THREADS)
void shortconv_kernel(const float* __restrict__ x,
                      const float* __restrict__ w,
                      float* __restrict__ out) {
    const int tid   = (int)threadIdx.x;
    const int chunk = (int)blockIdx.x;   // which 256-row sequence chunk
    const int dtile = (int)blockIdx.y;   // which 512-channel tile
    const int b     = (int)blockIdx.z;   // batch

    const int s0   = chunk * CH;
    const int dcol = dtile * DT + 4 * tid;        // first of this thread's 4 channels

    // Double-buffered LDS staging area: 2 stages x 8 rows x 512 ch x 4B = 32 KB
    __shared__ float smem[2 * TS * DT];
    const uint32_t lcol     = 4u * (uint32_t)tid;               // channel col in tile
    const uint32_t lds_base = (uint32_t)(uintptr_t)(&smem[0]);  // LDS byte address

    // Per-channel taps: w is D x 4 row-major, so one channel's taps = one float4.
    const float4* w4 = (const float4*)w;
    const float4 wr0 = w4[dcol + 0];
    const float4 wr1 = w4[dcol + 1];
    const float4 wr2 = w4[dcol + 2];
    const float4 wr3 = w4[dcol + 3];
    // Transpose to per-tap vectors across the 4 owned channels.
    const float4 wt0 = make_float4(wr0.x, wr1.x, wr2.x, wr3.x); // tap k=0 -> x[s-3]
    const float4 wt1 = make_float4(wr0.y, wr1.y, wr2.y, wr3.y); // tap k=1 -> x[s-2]
    const float4 wt2 = make_float4(wr0.z, wr1.z, wr2.z, wr3.z); // tap k=2 -> x[s-1]
    const float4 wt3 = make_float4(wr0.w, wr1.w, wr2.w, wr3.w); // tap k=3 -> x[s]

    // Sliding-window history: p1=x[s-1], p2=x[s-2], p3=x[s-3] (zeros before s=0).
    const float4 z4 = make_float4(0.f, 0.f, 0.f, 0.f);
    float4 p1 = z4, p2 = z4, p3 = z4;
    if (s0 - 1 >= 0)
        p1 = *(const float4*)(x + ((size_t)((size_t)b * S_ + (s0 - 1)) * D_ + dcol));
    if (s0 - 2 >= 0)
        p2 = *(const float4*)(x + ((size_t)((size_t)b * S_ + (s0 - 2)) * D_ + dcol));
    if (s0 - 3 >= 0)
        p3 = *(const float4*)(x + ((size_t)((size_t)b * S_ + (s0 - 3)) * D_ + dcol));

    // Async global->LDS staging (CDNA5 async data path, tracked by ASYNCcnt).
    const unsigned long long xbase = (unsigned long long)(uintptr_t)x;
    const uint32_t eoff0 =
        ((uint32_t)(b * S_ + s0)) * (uint32_t)D_ + (uint32_t)dcol; // element offset

    auto issue_stage = [&](int t) {
        const uint32_t lds_stage = lds_base + (uint32_t)((t & 1) * TS * DT * 4);
#pragma unroll
        for (int r = 0; r < TS; ++r) {
            const uint32_t voff  = (eoff0 + (uint32_t)((t * TS + r) * D_)) * 4u;
            const uint32_t laddr = lds_stage + (uint32_t)(r * DT * 4) + lcol * 4u;
            // Each lane DMAs its own 16B column slice into LDS (GVS mode:
            // 64-bit SGPR base + 32-bit per-lane byte offset).
            asm volatile("global_load_async_to_lds_b128 %0, %1, %2"
                         :: "v"(laddr), "v"(voff), "s"(xbase)
                         : "memory");
        }
    };

    issue_stage(0);
    issue_stage(1);

    for (int t = 0; t < NS; ++t) {
        // Async loads complete in order: with <=16 outstanding (stages t, t+1),
        // waiting down to 8 guarantees stage t's LDS writes have landed.
        if (t + 1 < NS) {
            asm volatile("s_wait_asynccnt 8" ::: "memory");
        } else {
            asm volatile("s_wait_asynccnt 0" ::: "memory");
        }

        const float* sbuf = &smem[(t & 1) * TS * DT];
#pragma unroll
        for (int r = 0; r < TS; ++r) {
            const float4 c = *(const float4*)(sbuf + r * DT + lcol);
            float4 o;
            o.x = fmaf(wt3.x, c.x, fmaf(wt2.x, p1.x, fmaf(wt1.x, p2.x, wt0.x * p3.x)));
            o.y = fmaf(wt3.y, c.y, fmaf(wt2.y, p1.y, fmaf(wt1.y, p2.y, wt0.y * p3.y)));
            o.z = fmaf(wt3.z, c.z, fmaf(wt2.z, p1.z, fmaf(wt1.z, p2.z, wt0.z * p3.z)));
            o.w = fmaf(wt3.w, c.w, fmaf(wt2.w, p1.w, fmaf(wt1.w, p2.w, wt0.w * p3.w)));

            const int s = s0 + t * TS + r;
            *(float4*)(out + ((size_t)((size_t)b * S_ + s) * D_ + dcol)) = o;

            p3 = p2; p2 = p1; p1 = c;
        }

        if (t + 2 < NS) issue_stage(t + 2);
    }
}

// new_cache = x[:, S-3:, :]  -> appended after `out` in d_out.
__global__ __launch_bounds__(256)
void shortconv_cache_kernel(const float* __restrict__ x, float* __restrict__ dst) {
    const int n4 = B_ * 3 * D_ / 4;
    const int i = (int)(blockIdx.x * blockDim.x + threadIdx.x);
    if (i >= n4) return;
    const int e = i * 4;
    const int d = e % D_;
    const int r = (e / D_) % 3;
    const int b = e / (3 * D_);
    const float4 v =
        *(const float4*)(x + ((size_t)((size_t)b * S_ + (S_ - 3 + r)) * D_ + d));
    *(float4*)(dst + ((size_t)(b * 3 + r) * D_ + d)) = v;
}

extern "C" void kernel_launch(void* const* d_in, const int* in_sizes, int n_in,
                              void* d_out, int out_size, void* d_ws, size_t ws_size,
                              hipStream_t stream) {
    (void)in_sizes; (void)n_in; (void)out_size; (void)d_ws; (void)ws_size;

    const float* x = (const float*)d_in[0];
    const float* w = (const float*)d_in[1];
    float* out   = (float*)d_out;
    float* cache = out + (size_t)B_ * S_ * D_;

    dim3 grid(S_ / CH, D_ / DT, B_);   // 16 x 4 x 8 = 512 blocks
    shortconv_kernel<<<grid, THREADS, 0, stream>>>(x, w, out);

    const int n4 = B_ * 3 * D_ / 4;
    shortconv_cache_kernel<<<(n4 + 255) / 256, 256, 0, stream>>>(x, cache);
}